// AttentionLayer_51221779972764
// MI455X (gfx1250) — compile-verified
//
#include <hip/hip_runtime.h>
#include <hip/hip_bf16.h>

#define B_  8
#define TE_ 2048
#define TD_ 256
#define H_  128

typedef __attribute__((ext_vector_type(2))) float v2f;
typedef __attribute__((ext_vector_type(8))) float v8f;

// ---------------------------------------------------------------------------
// Native CDNA5 tanh (V_TANH_F32 transcendental); exp2 fallback if unavailable.
// ---------------------------------------------------------------------------
__device__ __forceinline__ float tanh_fast(float x) {
#if __has_builtin(__builtin_amdgcn_tanhf)
    return __builtin_amdgcn_tanhf(x);
#else
    float t = __builtin_amdgcn_exp2f(-2.8853900817779268f * __builtin_fabsf(x));
    float r = (1.0f - t) / (1.0f + t);
    return __builtin_copysignf(r, x);
#endif
}

// ---------------------------------------------------------------------------
// fp32 GEMM via V_WMMA_F32_16X16X4_F32. One wave computes one 16x16 C tile.
// A: [M,K] row-major, Bm: [K,N] row-major, C: [M,N] row-major.
// Batched via simple pointer strides (grid.y = batch).
// A-matrix VGPR layout (ISA 7.12.2, 32-bit A 16x4): lanes 0-15 hold M=lane with
// K = k0,k0+1 in the two regs; lanes 16-31 hold K = k0+2,k0+3. B mirrors it.
// C/D layout: VGPR r -> M = r (lanes 0-15) / r+8 (lanes 16-31), N = lane&15.
// ---------------------------------------------------------------------------
__global__ void gemm_wmma_f32(const float* __restrict__ A,
                              const float* __restrict__ Bm,
                              float* __restrict__ C,
                              int M, int N, int K,
                              long aStride, long bStride, long cStride) {
    const int  batch = blockIdx.y;
    A  += (long)batch * aStride;
    Bm += (long)batch * bStride;
    C  += (long)batch * cStride;

    const int lane = threadIdx.x & 31;
    const int wave = blockIdx.x * (blockDim.x >> 5) + (threadIdx.x >> 5);
    const int ntn  = N >> 4;
    const int tm   = wave / ntn;
    const int tn   = wave - tm * ntn;
    if (tm >= (M >> 4)) return;

    const int m0 = tm << 4;
    const int n0 = tn << 4;
    const int mr = lane & 15;           // M row (A) / N col (B,C,D) within tile
    const int kk = (lane >> 4) << 1;    // upper half-wave handles K+2,K+3

    v8f acc = {};
    for (int k0 = 0; k0 < K; k0 += 4) {
        v2f a, b;
        const float* ap = A + (long)(m0 + mr) * K + (k0 + kk);
        a.x = ap[0];
        a.y = ap[1];
        const float* bp = Bm + (long)(k0 + kk) * N + (n0 + mr);
        b.x = bp[0];
        b.y = bp[N];
        acc = __builtin_amdgcn_wmma_f32_16x16x4_f32(
            /*neg_a=*/false, a, /*neg_b=*/false, b,
            /*c_mod=*/(short)0, acc, /*reuse_a=*/false, /*reuse_b=*/false);
    }

#pragma unroll
    for (int r = 0; r < 8; ++r) {
        int row = m0 + r + ((lane >> 4) << 3);
        C[(long)row * N + (n0 + mr)] = acc[r];
    }
}

// ---------------------------------------------------------------------------
// scores[b,j,t] = sum_k v[k] * tanh(Wenc[b,t,k] + Udec[b,j,k])
// Block = (batch b, tile of 8 j). 8 waves; wave w walks t = w, w+8, ...
// Lane holds k-chunk [4*lane, 4*lane+3] -> coalesced 512B Wenc row loads.
// Wave-wide sum via __shfl_xor (wave32).
// ---------------------------------------------------------------------------
__global__ void scores_kernel(const float* __restrict__ Wenc,  // [B,TE,H]
                              const float* __restrict__ Udec,  // [B,TD,H]
                              const float* __restrict__ V,     // [H]
                              float* __restrict__ S) {         // [B,TD,TE]
    const int b     = blockIdx.y;
    const int jBase = blockIdx.x << 3;          // 8 j per block
    const int lane  = threadIdx.x & 31;
    const int wave  = threadIdx.x >> 5;         // 0..7

    __shared__ __align__(16) float udec_s[8][H_];
    for (int i = threadIdx.x; i < 8 * H_; i += blockDim.x) {
        int j = i >> 7, k = i & (H_ - 1);
        udec_s[j][k] = Udec[((long)b * TD_ + jBase + j) * H_ + k];
    }
    const float4 v4 = ((const float4*)V)[lane];
    __syncthreads();

    const float* wb = Wenc + (long)b * TE_ * H_;
    float*       sb = S + ((long)b * TD_ + jBase) * TE_;

    for (int t = wave; t < TE_; t += 8) {
        const float4 we = ((const float4*)(wb + (long)t * H_))[lane];
#pragma unroll
        for (int j = 0; j < 8; ++j) {
            const float4 ud = ((const float4*)udec_s[j])[lane];
            float s = v4.x * tanh_fast(we.x + ud.x)
                    + v4.y * tanh_fast(we.y + ud.y)
                    + v4.z * tanh_fast(we.z + ud.z)
                    + v4.w * tanh_fast(we.w + ud.w);
#pragma unroll
            for (int off = 16; off > 0; off >>= 1)
                s += __shfl_xor(s, off, 32);
            if (lane == 0) sb[(long)j * TE_ + t] = s;
        }
    }
}

// ---------------------------------------------------------------------------
// In-place row softmax over TE. One 256-thread block per (b,j) row.
// ---------------------------------------------------------------------------
__global__ void softmax_kernel(float* __restrict__ S) {        // [B*TD, TE]
    float* p = S + (long)blockIdx.x * TE_;
    const int tid = threadIdx.x;
    __shared__ float red[256];

    float m = -INFINITY;
    for (int i = tid; i < TE_; i += 256) m = fmaxf(m, p[i]);
    red[tid] = m;
    __syncthreads();
    for (int s = 128; s > 0; s >>= 1) {
        if (tid < s) red[tid] = fmaxf(red[tid], red[tid + s]);
        __syncthreads();
    }
    m = red[0];
    __syncthreads();

    float sum = 0.0f;
    for (int i = tid; i < TE_; i += 256) {
        float e = __expf(p[i] - m);
        p[i] = e;
        sum += e;
    }
    red[tid] = sum;
    __syncthreads();
    for (int s = 128; s > 0; s >>= 1) {
        if (tid < s) red[tid] += red[tid + s];
        __syncthreads();
    }
    const float inv = 1.0f / red[0];
    for (int i = tid; i < TE_; i += 256) p[i] *= inv;
}

// ---------------------------------------------------------------------------
extern "C" void kernel_launch(void* const* d_in, const int* in_sizes, int n_in,
                              void* d_out, int out_size, void* d_ws, size_t ws_size,
                              hipStream_t stream) {
    const float* enc = (const float*)d_in[0];   // [B,TE,H]
    const float* dec = (const float*)d_in[1];   // [B,TD,D]
    const float* W_a = (const float*)d_in[2];   // [H,H]
    const float* U_a = (const float*)d_in[3];   // [D,H]
    const float* V_a = (const float*)d_in[4];   // [H,1]

    float* c_out = (float*)d_out;                          // [B,TD,H]
    float* e_out = c_out + (long)B_ * TD_ * H_;            // [B,TD,TE]

    float* Wenc = (float*)d_ws;                            // [B,TE,H]
    float* Udec = Wenc + (long)B_ * TE_ * H_;              // [B,TD,H]

    // 1) Wenc = enc @ W_a   (M = B*TE = 16384, N = K = 128)
    {
        int M = B_ * TE_, N = H_, K = H_;
        int waves = (M >> 4) * (N >> 4);                   // 8192
        gemm_wmma_f32<<<dim3(waves / 8, 1), 256, 0, stream>>>(
            enc, W_a, Wenc, M, N, K, 0, 0, 0);
    }
    // 2) Udec = dec @ U_a   (M = B*TD = 2048)
    {
        int M = B_ * TD_, N = H_, K = H_;
        int waves = (M >> 4) * (N >> 4);                   // 1024
        gemm_wmma_f32<<<dim3(waves / 8, 1), 256, 0, stream>>>(
            dec, U_a, Udec, M, N, K, 0, 0, 0);
    }
    // 3) raw scores -> e_out region
    scores_kernel<<<dim3(TD_ / 8, B_), 256, 0, stream>>>(Wenc, Udec, V_a, e_out);

    // 4) softmax rows in place
    softmax_kernel<<<dim3(B_ * TD_), 256, 0, stream>>>(e_out);

    // 5) c[b] = e[b] @ enc[b]   (M=TD=256, N=H=128, K=TE=2048, batched)
    {
        int M = TD_, N = H_, K = TE_;
        int wavesPerBatch = (M >> 4) * (N >> 4);           // 128
        gemm_wmma_f32<<<dim3(wavesPerBatch / 8, B_), 256, 0, stream>>>(
            e_out, enc, c_out, M, N, K,
            (long)TD_ * TE_, (long)TE_ * H_, (long)TD_ * H_);
    }
}